// MinibatchDiscrimination_5806795784869
// MI455X (gfx1250) — compile-verified
//
#include <hip/hip_runtime.h>

typedef __attribute__((ext_vector_type(2))) float v2f;
typedef __attribute__((ext_vector_type(8))) float v8f;

#define N_SAMP 256
#define IN_F   512
#define OUT_F  128
#define KD     16
#define MDIM   (OUT_F * KD)   // 2048
#define OUT_W  (IN_F + OUT_F) // 640

// ---------------------------------------------------------------------------
// Kernel 1: M[256,2048] = x[256,512] @ T[512,2048] using V_WMMA_F32_16X16X4_F32
// One wave32 computes one 16x16 tile of M; K loop advances 4 per WMMA.
//
// VGPR layouts (CDNA5 ISA 7.12.2):
//   A (16x4 f32, 2 VGPRs): lane L holds A[L&15][khalf+0], A[L&15][khalf+1]
//                          where khalf = (L>=16) ? 2 : 0
//   B (4x16 f32, 2 VGPRs): VGPR r holds B[khalf+r][lane&15]
//   C/D (16x16 f32, 8 VGPRs): VGPR r -> row r + (lane>=16 ? 8 : 0), col lane&15
// ---------------------------------------------------------------------------
__global__ void __launch_bounds__(128)
mbd_gemm_wmma_f32(const float* __restrict__ x, const float* __restrict__ T,
                  float* __restrict__ M) {
  const int lane      = threadIdx.x & 31;
  const int waveInBlk = threadIdx.x >> 5;
  const int waveId    = blockIdx.x * 4 + waveInBlk;  // 0..2047
  const int ntilesN   = MDIM / 16;                   // 128
  const int tm        = (waveId / ntilesN) * 16;     // output row base
  const int tn        = (waveId % ntilesN) * 16;     // output col base

  const int rc    = lane & 15;               // row (for A) / col (for B,C,D)
  const int khalf = (lane >> 4) << 1;        // 0 or 2

  const float* __restrict__ arow = x + (tm + rc) * IN_F + khalf; // A[row][khalf+...]
  const float* __restrict__ bcol = T + khalf * MDIM + tn + rc;   // B[khalf+...][col]

  v8f c = {};

  #pragma unroll 4
  for (int k = 0; k < IN_F; k += 4) {
    // A fragment: contiguous float2, 8-byte aligned (khalf even, k % 4 == 0)
    v2f a = *(const v2f*)(arow + k);
    // B fragment: two rows of T, stride MDIM
    v2f b;
    b.x = bcol[(size_t)k * MDIM];
    b.y = bcol[(size_t)(k + 1) * MDIM];
    // (neg_a, A, neg_b, B, c_mod, C, reuse_a, reuse_b)
    c = __builtin_amdgcn_wmma_f32_16x16x4_f32(false, a, false, b,
                                              (short)0, c, false, false);
  }

  // Scatter D back: VGPR r -> M[tm + r + (lane>=16?8:0)][tn + (lane&15)]
  const int rbase = tm + ((lane >> 4) << 3);
  float* __restrict__ dst = M + (size_t)rbase * MDIM + tn + rc;
  #pragma unroll
  for (int r = 0; r < 8; ++r) {
    dst[(size_t)r * MDIM] = c[r];
  }
}

// ---------------------------------------------------------------------------
// Kernel 2: pairwise L1 -> exp(-d) -> batch reduce, plus copy of x into out.
// Grid: 256 blocks (one per sample j), 128 threads (one per output feature o).
// out row j = [ x[j, 0:512] | o_b[j, 0:128] ],  o_b[j,o] =
//     sum_i exp( -sum_k |M[i,o,k] - M[j,o,k]| ) - 1
// M (2 MB) stays L2-resident across all 256 blocks; compute is VALU/exp bound.
// ---------------------------------------------------------------------------
__global__ void __launch_bounds__(128)
mbd_pairwise_exp(const float* __restrict__ x, const float* __restrict__ M,
                 float* __restrict__ out) {
  const int j = blockIdx.x;    // sample
  const int o = threadIdx.x;   // output feature 0..127

  // Copy x row into the first IN_F columns of the output row.
  const float* __restrict__ xr   = x + (size_t)j * IN_F;
  float* __restrict__       orow = out + (size_t)j * OUT_W;
  #pragma unroll
  for (int c = o; c < IN_F; c += OUT_F) orow[c] = xr[c];

  // My reference chunk M[j, o, 0:16] in registers.
  float mj[KD];
  const float* __restrict__ mjp = M + (size_t)j * MDIM + o * KD;
  #pragma unroll
  for (int k = 0; k < KD; ++k) mj[k] = mjp[k];

  float acc = 0.0f;
  for (int i = 0; i < N_SAMP; ++i) {
    const float4* __restrict__ mip =
        (const float4*)(M + (size_t)i * MDIM + o * KD);
    float d = 0.0f;
    #pragma unroll
    for (int q = 0; q < 4; ++q) {
      float4 v = mip[q];
      d += fabsf(v.x - mj[4 * q + 0]);
      d += fabsf(v.y - mj[4 * q + 1]);
      d += fabsf(v.z - mj[4 * q + 2]);
      d += fabsf(v.w - mj[4 * q + 3]);
    }
    acc += __expf(-d);   // v_exp_f32 path
  }
  orow[IN_F + o] = acc - 1.0f;  // -1 removes self-similarity (i == j term)
}

// ---------------------------------------------------------------------------
extern "C" void kernel_launch(void* const* d_in, const int* in_sizes, int n_in,
                              void* d_out, int out_size, void* d_ws, size_t ws_size,
                              hipStream_t stream) {
  const float* x = (const float*)d_in[0];   // [256, 512]
  const float* T = (const float*)d_in[1];   // [512, 128, 16] == [512, 2048]
  float*       out = (float*)d_out;         // [256, 640]
  float*       M   = (float*)d_ws;          // [256, 2048] scratch (2 MB)

  // 2048 tiles of 16x16, 4 waves per block -> 512 blocks.
  mbd_gemm_wmma_f32<<<512, 128, 0, stream>>>(x, T, M);
  // One block per sample, one thread per output feature.
  mbd_pairwise_exp<<<N_SAMP, OUT_F, 0, stream>>>(x, M, out);
}